// RenderRay_63479616635279
// MI455X (gfx1250) — compile-verified
//
#include <hip/hip_runtime.h>
#include <hip/hip_bf16.h>
#include <math.h>
#include <stdint.h>

// ---------------------------------------------------------------------------
// NeRF hierarchical sampling + volume rendering (color_map only).
//   B=65536 rays, NC=64 coarse, NF=128 fine, NS=192 composite samples.
// One wave32 per ray; 8 waves (8 rays) per 256-thread block.
// Density/color staged global->LDS with CDNA5 async-to-LDS copies overlapped
// with the CDF/sampling/sort phases.
// ---------------------------------------------------------------------------

#define B_RAYS 65536
#define NCOARSE 64
#define NFINE 128
#define NS 192          // NCOARSE + NFINE
#define MW 62           // weights actually used: coarse_weights[:,1:-1]
#define RPB 8           // rays (waves) per block
#define TPB (RPB * 32)

#if defined(__HIP_DEVICE_COMPILE__) && defined(__gfx1250__) && \
    __has_builtin(__builtin_amdgcn_global_load_async_to_lds_b128)
#define ASYNC_LDS 1
#else
#define ASYNC_LDS 0
#endif

#define GAS __attribute__((address_space(1)))
#define LAS __attribute__((address_space(3)))

typedef int v4i32 __attribute__((__vector_size__(16)));

__device__ __forceinline__ void async_copy16(const float* g, float* l) {
#if ASYNC_LDS
  // Builtin param 1 is v4i in AS(1) (global); param 2 is the LDS destination.
  // Flat LDS address truncates to the LDS byte offset (ISA 10.2 aperture
  // table: the low 32 address bits are kept); global flat == AS1 numerically,
  // so integer round-trips give the correct per-AS pointers.
  __builtin_amdgcn_global_load_async_to_lds_b128(
      (GAS v4i32*)(uintptr_t)g,
      (LAS v4i32*)(unsigned int)(uintptr_t)l,
      0, 0);
#else
  (void)g; (void)l;
#endif
}

__device__ __forceinline__ void wait_async_lds() {
#if ASYNC_LDS
#if __has_builtin(__builtin_amdgcn_s_wait_asynccnt)
  __builtin_amdgcn_s_wait_asynccnt(0);
#else
  asm volatile("s_wait_asynccnt 0" ::: "memory");
#endif
#endif
}

__device__ __forceinline__ float wave_reduce_add(float x) {
#pragma unroll
  for (int off = 16; off > 0; off >>= 1) x += __shfl_xor(x, off, 32);
  return x;
}

__device__ __forceinline__ float wave_scan_incl_add(float x, int lane) {
#pragma unroll
  for (int off = 1; off < 32; off <<= 1) {
    float t = __shfl_up(x, off, 32);
    if (lane >= off) x += t;
  }
  return x;
}

__device__ __forceinline__ float wave_scan_incl_mul(float x, int lane) {
#pragma unroll
  for (int off = 1; off < 32; off <<= 1) {
    float t = __shfl_up(x, off, 32);
    if (lane >= off) x *= t;
  }
  return x;
}

__global__ __launch_bounds__(TPB) void nerf_fine_render_kernel(
    const float* __restrict__ cdepth,   // [B,64]
    const float* __restrict__ cweight,  // [B,64]
    const float* __restrict__ usamp,    // [B,128]
    const float* __restrict__ gdens,    // [B,192]
    const float* __restrict__ gcol,     // [B,192,3]
    float* __restrict__ out)            // [B,3]
{
  __shared__ __align__(16) float sSort[RPB][256];   // 64 coarse + 128 fine + 64 INF pad
  __shared__ __align__(16) float sDens[RPB][NS];
  __shared__ __align__(16) float sCol[RPB][NS * 3];
  __shared__ __align__(16) float sCdf[RPB][64];     // 63 used
  __shared__ __align__(16) float sMids[RPB][64];    // 63 used

  const int lane = threadIdx.x & 31;
  const int w    = threadIdx.x >> 5;
  const int ray  = blockIdx.x * RPB + w;   // grid sized exactly: no guard needed

  const float* gD = gdens + (size_t)ray * NS;
  const float* gC = gcol  + (size_t)ray * NS * 3;

  // ---- Phase A: stage density (48 x float4) + color (144 x float4) to LDS ----
#if ASYNC_LDS
  for (int i = lane; i < NS / 4; i += 32)        async_copy16(gD + 4 * i, &sDens[w][4 * i]);
  for (int i = lane; i < (NS * 3) / 4; i += 32)  async_copy16(gC + 4 * i, &sCol[w][4 * i]);
#else
  for (int i = lane; i < NS / 4; i += 32)
    *(float4*)&sDens[w][4 * i] = *(const float4*)(gD + 4 * i);
  for (int i = lane; i < (NS * 3) / 4; i += 32)
    *(float4*)&sCol[w][4 * i] = *(const float4*)(gC + 4 * i);
#endif

  // ---- coarse depths into sort buffer; pad tail with +INF ----
  const float2 cd2 = *(const float2*)(cdepth + (size_t)ray * NCOARSE + 2 * lane);
  sSort[w][2 * lane]     = cd2.x;
  sSort[w][2 * lane + 1] = cd2.y;
  sSort[w][NS + 2 * lane]     = __builtin_huge_valf();
  sSort[w][NS + 2 * lane + 1] = __builtin_huge_valf();
  __syncthreads();

  // ---- Phase B: mids + pdf/cdf ----
  {
    const int j0 = 2 * lane, j1 = 2 * lane + 1;
    if (j0 < 63) sMids[w][j0] = 0.5f * (sSort[w][j0] + sSort[w][j0 + 1]);
    if (j1 < 63) sMids[w][j1] = 0.5f * (sSort[w][j1] + sSort[w][j1 + 1]);
  }
  {
    const float* gW = cweight + (size_t)ray * NCOARSE;
    const int e0 = 2 * lane, e1 = 2 * lane + 1;    // element e maps to coarse index e+1
    const float p0 = (e0 < MW) ? (gW[e0 + 1] + 1e-5f) : 0.0f;
    const float p1 = (e1 < MW) ? (gW[e1 + 1] + 1e-5f) : 0.0f;
    const float tot  = wave_reduce_add(p0 + p1);
    const float pdf0 = p0 / tot;
    const float pdf1 = p1 / tot;
    const float S = wave_scan_incl_add(pdf0 + pdf1, lane);  // inclusive through e1
    if (lane == 0) sCdf[w][0] = 0.0f;
    if (e0 < MW) sCdf[w][e0 + 1] = S - pdf1;
    if (e1 < MW) sCdf[w][e1 + 1] = S;
  }
  __syncthreads();

  // ---- Phase C: inverse-CDF sampling, 4 u's per lane ----
  float fines[4];
#pragma unroll
  for (int r = 0; r < 4; ++r) {
    const float u = usamp[(size_t)ray * NFINE + lane + 32 * r];
    // searchsorted(cdf[0:62], u, 'right') -> above in [0,62]
    int lo = 0, hi = MW;
    while (lo < hi) {
      const int mid = (lo + hi) >> 1;
      if (sCdf[w][mid] <= u) lo = mid + 1; else hi = mid;
    }
    const int above = lo;
    const int below = (above > 0) ? above - 1 : 0;
    const float cb = sCdf[w][below], ca = sCdf[w][above];
    const float bb = sMids[w][below], ba = sMids[w][above];
    float den = ca - cb;
    if (den < 1e-5f) den = 1.0f;
    const float t = (u - cb) / den;
    fines[r] = bb + t * (ba - bb);
  }
#pragma unroll
  for (int r = 0; r < 4; ++r) sSort[w][NCOARSE + lane + 32 * r] = fines[r];
  __syncthreads();

  // ---- Phase D: bitonic sort of 256 (192 real + INF pad) ----
  for (int k = 2; k <= 256; k <<= 1) {
    for (int j = k >> 1; j > 0; j >>= 1) {
      for (int t = lane; t < 128; t += 32) {
        const int i   = ((t & ~(j - 1)) << 1) | (t & (j - 1));
        const int ixj = i | j;
        const float a = sSort[w][i];
        const float b = sSort[w][ixj];
        const bool up = ((i & k) == 0);
        if (up ? (a > b) : (a < b)) { sSort[w][i] = b; sSort[w][ixj] = a; }
      }
      __syncthreads();
    }
  }

  // ---- Phase E: composite (6 consecutive samples per lane) ----
  wait_async_lds();
  const int base = 6 * lane;
  float L[6], A[6];
  float P = 1.0f;
#pragma unroll
  for (int i = 0; i < 6; ++i) {
    const int g = base + i;
    const float d = sSort[w][g];
    const float dist = (g == NS - 1) ? 1e10f : (sSort[w][g + 1] - d);
    const float alpha = 1.0f - expf(-sDens[w][g] * dist);
    L[i] = P;                       // local exclusive prefix of transmittance factors
    A[i] = alpha;
    P *= (1.0f - alpha + 1e-10f);
  }
  const float incl = wave_scan_incl_mul(P, lane);
  float ex = __shfl_up(incl, 1, 32);
  if (lane == 0) ex = 1.0f;

  float c0 = 0.0f, c1 = 0.0f, c2 = 0.0f;
#pragma unroll
  for (int i = 0; i < 6; ++i) {
    const int g = base + i;
    const float wgt = ex * L[i] * A[i];
    c0 += wgt * sCol[w][3 * g];
    c1 += wgt * sCol[w][3 * g + 1];
    c2 += wgt * sCol[w][3 * g + 2];
  }
  c0 = wave_reduce_add(c0);
  c1 = wave_reduce_add(c1);
  c2 = wave_reduce_add(c2);
  if (lane == 0) {
    out[3 * (size_t)ray]     = c0;
    out[3 * (size_t)ray + 1] = c1;
    out[3 * (size_t)ray + 2] = c2;
  }
}

extern "C" void kernel_launch(void* const* d_in, const int* in_sizes, int n_in,
                              void* d_out, int out_size, void* d_ws, size_t ws_size,
                              hipStream_t stream) {
  (void)in_sizes; (void)n_in; (void)out_size; (void)d_ws; (void)ws_size;
  // inputs: 0 ray_origin (unused), 1 ray_direction (unused), 2 coarse_depth_values,
  //         3 coarse_weights, 4 u, 5 raw_density, 6 raw_color
  const float* cdepth  = (const float*)d_in[2];
  const float* cweight = (const float*)d_in[3];
  const float* usamp   = (const float*)d_in[4];
  const float* gdens   = (const float*)d_in[5];
  const float* gcol    = (const float*)d_in[6];
  float* out = (float*)d_out;

  dim3 grid(B_RAYS / RPB);
  dim3 block(TPB);
  nerf_fine_render_kernel<<<grid, block, 0, stream>>>(cdepth, cweight, usamp,
                                                      gdens, gcol, out);
}